// GeometricPoseEstimator_58944131170647
// MI455X (gfx1250) — compile-verified
//
#include <hip/hip_runtime.h>
#include <math.h>

// Problem constants (match reference)
#define NB 8
#define NN 512
#define NM 512
#define NH 64
#define MT 64      // m-columns handled per block (NM/8)
#define NBUCK 2048 // bucket resolution over [0,1)
#define SCAN 4     // compile-time scan window for the fast lookup path

typedef __attribute__((ext_vector_type(2))) float v2f;
typedef __attribute__((ext_vector_type(8))) float v8f;

// Workspace layout (floats):
//   ws[0..7]            : per-batch max of match_scores
//   ws[8..8+NB*17)      : per-batch accumulators (see below)
//   ws[WS_ST..+128)     : sortedT[128] (sorted ReLU breakpoints, +INF padded)
//   ws[WS_SA..+65)      : segA[65]     (piecewise-linear slopes of z(s))
//   ws[WS_SB..+65)      : segB[65]     (piecewise-linear intercepts of z(s))
//   ws[WS_CNT..+NBUCK)  : cnt0[NBUCK]  (#breakpoints < i/NBUCK, as float 0..64)
//   ws[WS_MAXOCC]       : maxocc       (max #breakpoints in any bucket)
// Accumulators per batch: a[0]=Wtot, a[1..3]=c1u, a[4..6]=c2u,
//                         a[7..15]=G' (row-major d,e), a[16]=count
#define ACC_STRIDE 17
#define WS_MAXS 0
#define WS_ACC 8
#define WS_ST (8 + NB * ACC_STRIDE)   // 144
#define WS_SA (WS_ST + 128)           // 272
#define WS_SB (WS_SA + 65)            // 337
#define WS_CNT (WS_SB + 65)           // 402
#define WS_MAXOCC (WS_CNT + NBUCK)    // 2450

// ---------------------------------------------------------------------------
// Kernel 0: build the piecewise-linear representation of the weight-net:
//   z(s) = b2 + sum_j w2_j * relu(w1_j*s + b1_j)  ==  A_k*s + B_k  on segment k,
// plus a 2048-bucket index over [0,1) so the hot-kernel segment lookup is
// one ds_load_u8 + a fixed 4-wide window of independent compares.
// ---------------------------------------------------------------------------
__global__ void __launch_bounds__(64)
k_prepare(const float* __restrict__ w1, const float* __restrict__ b1,
          const float* __restrict__ w2, const float* __restrict__ b2,
          float* __restrict__ ws) {
  __shared__ float key[64], sT[64], sdA[64], sdB[64], cA[64], cB[64];
  __shared__ int occmax[64];
  const int j = threadIdx.x;  // 64 threads
  const float a = w1[j], bb = b1[j], c = w2[j];

  // breakpoint & activation delta when s crosses it upward
  float t, da, db;
  if (a > 0.0f)      { t = -bb / a; da =  a * c; db =  bb * c; }   // activates
  else if (a < 0.0f) { t = -bb / a; da = -a * c; db = -bb * c; }   // deactivates
  else               { t = 3.0e38f; da = 0.0f;   db = 0.0f;   }    // no breakpoint

  // contribution to the s -> -inf segment
  cA[j] = (a < 0.0f) ? a * c : 0.0f;
  cB[j] = (a < 0.0f) ? bb * c : ((a == 0.0f && bb > 0.0f) ? bb * c : 0.0f);

  key[j] = t;
  __syncthreads();
  int rank = 0;
  for (int k = 0; k < 64; ++k) {
    const float tk = key[k];
    rank += (tk < t || (tk == t && k < j)) ? 1 : 0;
  }
  sT[rank] = t; sdA[rank] = da; sdB[rank] = db;

  // +INF padding for indices 64..127 (scan sentinels)
  ws[WS_ST + 64 + j] = 3.0e38f;
  __syncthreads();

  if (j == 0) {
    float A = 0.0f, Bv = b2[0];
    for (int k = 0; k < 64; ++k) { A += cA[k]; Bv += cB[k]; }
    float* sortedT = ws + WS_ST;
    float* segA = ws + WS_SA;
    float* segB = ws + WS_SB;
    segA[0] = A; segB[0] = Bv;
    for (int k = 0; k < 64; ++k) {
      sortedT[k] = sT[k];
      A += sdA[k]; Bv += sdB[k];
      segA[k + 1] = A; segB[k + 1] = Bv;
    }
  }

  // Bucket index over [0,1): cnt0[i] = #(t < i/NBUCK); occ = per-bucket count.
  // i/NBUCK is exact in fp32 (power-of-two denominator).
  int locmax = 0;
  for (int i = j; i < NBUCK; i += 64) {
    const float lo = (float)i * (1.0f / (float)NBUCK);
    const float hi = (i == NBUCK - 1) ? 1.0f : (float)(i + 1) * (1.0f / (float)NBUCK);
    int clo = 0, chi = 0;
    for (int k = 0; k < 64; ++k) {
      clo += (sT[k] < lo) ? 1 : 0;
      chi += (sT[k] < hi) ? 1 : 0;
    }
    ws[WS_CNT + i] = (float)clo;
    locmax = (chi - clo) > locmax ? (chi - clo) : locmax;
  }
  occmax[j] = locmax;
  __syncthreads();
  if (j == 0) {
    int mo = 0;
    for (int k = 0; k < 64; ++k) mo = occmax[k] > mo ? occmax[k] : mo;
    ws[WS_MAXOCC] = (float)mo;
  }
}

// ---------------------------------------------------------------------------
// Kernel 1: per-batch max reduction + zero accumulators (graph-replay safe).
// ---------------------------------------------------------------------------
__global__ void k_max_init(const float* __restrict__ scores,
                           float* __restrict__ max_s,
                           float* __restrict__ acc) {
  const int b = blockIdx.x;
  const float* S = scores + (size_t)b * NN * NM;
  float m = -1e30f;
  for (int i = threadIdx.x; i < NN * NM; i += 256)
    m = fmaxf(m, S[i]);
  __shared__ float red[256];
  red[threadIdx.x] = m;
  __syncthreads();
  for (int off = 128; off > 0; off >>= 1) {
    if (threadIdx.x < off)
      red[threadIdx.x] = fmaxf(red[threadIdx.x], red[threadIdx.x + off]);
    __syncthreads();
  }
  if (threadIdx.x == 0) max_s[b] = red[0];
  if (threadIdx.x < ACC_STRIDE) acc[b * ACC_STRIDE + threadIdx.x] = 0.0f;
}

// ---------------------------------------------------------------------------
// Hot loop body, instantiated twice (wave-uniform selection, no divergence):
//   FAST = true : bucket base (ds_load_u8) + fixed SCAN-wide window of
//                 independent compares (exact iff every bucket holds <= SCAN
//                 breakpoints; guaranteed by the caller's maxocc guard).
//   FAST = false: universally exact 7-step branchless binary search.
// All loop bounds are compile-time constants -> straight-line codegen,
// no loop versioning.
// ---------------------------------------------------------------------------
template <bool FAST>
__device__ __forceinline__ void cov_loop(
    const float* __restrict__ S, const float* sT, const float* sA,
    const float* sB, const unsigned char* cT8, const float* sp2,
    float thr, int lane, v8f& cacc, float& rowsum, float& cntf) {
  const int row   = lane & 15;
  const int half  = lane >> 4;
  const int kbase = half * 2;
  const int colB  = lane & 15;

#pragma unroll 2
  for (int m0 = 0; m0 < MT; m0 += 4) {
    // two contiguous scores for this lane's (row, K) slots: 8B-aligned load
    const float2 sv = *(const float2*)(S + (size_t)row * NM + m0 + kbase);
    const float ssv[2] = {sv.x, sv.y};
    float wp[2];
#pragma unroll
    for (int t = 0; t < 2; ++t) {
      const float s = ssv[t];
      int k;
      if (FAST) {
        int ib = (int)(s * (float)NBUCK);
        ib = ib < 0 ? 0 : (ib > NBUCK - 1 ? NBUCK - 1 : ib);
        const int kb = (int)cT8[ib];
        k = kb;
#pragma unroll
        for (int i = 0; i < SCAN; ++i)
          k += (sT[kb + i] <= s) ? 1 : 0;     // independent loads, +INF pads
      } else {
        k = 0;
#pragma unroll
        for (int stepv = 64; stepv > 0; stepv >>= 1)
          k += (sT[k + stepv - 1] <= s) ? stepv : 0;
      }
      const float z = fmaf(sA[k], s, sB[k]);
      const float sg = __builtin_amdgcn_rcpf(1.0f + __expf(-z));  // sigmoid
      const bool msk = s > thr;
      wp[t] = msk ? sg : 0.0f;
      cntf += msk ? 1.0f : 0.0f;
    }
    rowsum += wp[0] + wp[1];

    v2f a; a.x = wp[0]; a.y = wp[1];
    v2f bbv;
    bbv.x = (colB < 3) ? sp2[(m0 + kbase) * 3 + colB] : 0.0f;
    bbv.y = (colB < 3) ? sp2[(m0 + kbase + 1) * 3 + colB] : 0.0f;

    cacc = __builtin_amdgcn_wmma_f32_16x16x4_f32(
        /*neg_a=*/false, a, /*neg_b=*/false, bbv,
        /*c_mod=*/(short)0, cacc, /*reuse_a=*/false, /*reuse_b=*/false);
  }
}

// ---------------------------------------------------------------------------
// Kernel 2: one wave per (batch, 16-row tile of N, 1/8 slice of M).
// Weights go directly into the V_WMMA_F32_16X16X4_F32 A-matrix layout,
// contracting W_tile x pos2 with chained fp32 WMMAs (C carried across M loop).
// A 16x4 f32: lanes 0-15 -> (row=lane, K=0/1), lanes 16-31 -> (row, K=2/3).
// B 4x16 f32: v0 = rows {K=0|K=2}, v1 = rows {K=1|K=3}, col = lane&15
//             (cols 0..2 = pos2 components, rest zero).
// C 16x16 f32: VGPR v holds (M=v, N=lane) / (M=v+8, N=lane-16).
// ---------------------------------------------------------------------------
__global__ void __launch_bounds__(32)
k_weights_cov(const float* __restrict__ pos1, const float* __restrict__ pos2,
              const float* __restrict__ scores,
              const float* __restrict__ max_s, const float* __restrict__ ws,
              float* __restrict__ acc) {
  __shared__ float sT[128], sA[65], sB[65];
  __shared__ unsigned char cT8[NBUCK];
  __shared__ float sp2[MT * 3];

  const int lane = threadIdx.x;
  const int bi = blockIdx.x;
  const int b = bi >> 8;             // 32 tiles * 8 m-slices per batch
  const int tile = (bi >> 3) & 31;
  const int mq = bi & 7;
  const int n0 = tile * 16;
  const int m0base = mq * MT;

  for (int j = lane; j < 128; j += 32) sT[j] = ws[WS_ST + j];
  for (int j = lane; j < 65; j += 32) { sA[j] = ws[WS_SA + j]; sB[j] = ws[WS_SB + j]; }
  for (int j = lane; j < NBUCK; j += 32) cT8[j] = (unsigned char)ws[WS_CNT + j];
  const float* P2 = pos2 + (size_t)b * NM * 3;
  for (int j = lane; j < MT * 3; j += 32) sp2[j] = P2[m0base * 3 + j];
  __syncthreads();

  const int maxocc = (int)ws[WS_MAXOCC];   // wave-uniform
  const float thr = 0.1f * max_s[b];
  const float* S  = scores + (size_t)b * NN * NM + (size_t)n0 * NM + m0base;
  const float* P1 = pos1 + (size_t)b * NN * 3;

  v8f cacc = {};
  float rowsum = 0.0f;
  float cntf = 0.0f;

  if (maxocc <= SCAN) {
    cov_loop<true>(S, sT, sA, sB, cT8, sp2, thr, lane, cacc, rowsum, cntf);
  } else {
    cov_loop<false>(S, sT, sA, sB, cT8, sp2, thr, lane, cacc, rowsum, cntf);
  }

  const int row  = lane & 15;
  const int half = lane >> 4;
  const int colB = lane & 15;
  float* Aa = acc + b * ACC_STRIDE;

  // ---- wave-reduced accumulation (17 atomics per block) ----
  float c = cntf;
  for (int off = 16; off > 0; off >>= 1) c += __shfl_xor(c, off, 32);
  // wn per row (both halves end up with the full row sum -> halve when summing
  // over all 32 lanes)
  const float wn = rowsum + __shfl_xor(rowsum, 16, 32);
  const float* p1r = P1 + (size_t)(n0 + row) * 3;
  float v0 = 0.5f * wn;
  float v1 = 0.5f * wn * p1r[0];
  float v2 = 0.5f * wn * p1r[1];
  float v3 = 0.5f * wn * p1r[2];
  for (int off = 16; off > 0; off >>= 1) {
    v0 += __shfl_xor(v0, off, 32);
    v1 += __shfl_xor(v1, off, 32);
    v2 += __shfl_xor(v2, off, 32);
    v3 += __shfl_xor(v3, off, 32);
  }
  if (lane == 0) {
    atomicAdd(&Aa[16], c);
    atomicAdd(&Aa[0], v0);
    atomicAdd(&Aa[1], v1);
    atomicAdd(&Aa[2], v2);
    atomicAdd(&Aa[3], v3);
  }

  // c2u[e] += sum_n r[n][e] ;  G'[d][e] += sum_n pos1[n][d]*r[n][e]
  float rsum = 0.0f, g0 = 0.0f, g1 = 0.0f, g2 = 0.0f;
  if (colB < 3) {
#pragma unroll
    for (int v = 0; v < 8; ++v) {
      const int rr = v + half * 8;
      const float rv = cacc[v];
      rsum += rv;
      const float* p1q = P1 + (size_t)(n0 + rr) * 3;
      g0 = fmaf(p1q[0], rv, g0);
      g1 = fmaf(p1q[1], rv, g1);
      g2 = fmaf(p1q[2], rv, g2);
    }
  }
  rsum += __shfl_xor(rsum, 16, 32);
  g0 += __shfl_xor(g0, 16, 32);
  g1 += __shfl_xor(g1, 16, 32);
  g2 += __shfl_xor(g2, 16, 32);
  if (lane < 3) {
    atomicAdd(&Aa[4 + colB], rsum);
    atomicAdd(&Aa[7 + 0 * 3 + colB], g0);
    atomicAdd(&Aa[7 + 1 * 3 + colB], g1);
    atomicAdd(&Aa[7 + 2 * 3 + colB], g2);
  }
}

// ---------------------------------------------------------------------------
// Kernel 3: per-batch Kabsch finalize (8 lanes, one batch each).
// H = G'/T - (2 - S/T) c1 c2^T ;  SVD via Jacobi on H^T H ;  R = V diag(1,1,d) U^T
// ---------------------------------------------------------------------------
__global__ void k_finalize(const float* __restrict__ acc, float* __restrict__ out) {
  const int b = threadIdx.x;
  if (b >= NB) return;
  const float* Aa = acc + b * ACC_STRIDE;

  const float Sw = Aa[0];
  const float T = Sw + 1e-8f;
  const float invT = 1.0f / T;
  float c1[3], c2[3];
  for (int d = 0; d < 3; ++d) { c1[d] = Aa[1 + d] * invT; c2[d] = Aa[4 + d] * invT; }
  const float fac = 2.0f - Sw * invT;

  float H[3][3];
  for (int d = 0; d < 3; ++d)
    for (int e = 0; e < 3; ++e)
      H[d][e] = Aa[7 + d * 3 + e] * invT - fac * c1[d] * c2[e];

  // Symmetric Am = H^T H; Jacobi eigendecomposition -> V (columns), lam
  float Am[3][3];
  for (int i = 0; i < 3; ++i)
    for (int j = 0; j < 3; ++j) {
      float s = 0.0f;
      for (int k = 0; k < 3; ++k) s += H[k][i] * H[k][j];
      Am[i][j] = s;
    }
  float V[3][3] = {{1,0,0},{0,1,0},{0,0,1}};
  for (int sweep = 0; sweep < 12; ++sweep) {
    for (int pair = 0; pair < 3; ++pair) {
      const int p = (pair == 2) ? 1 : 0;
      const int q = (pair == 0) ? 1 : 2;
      const float apq = Am[p][q];
      if (fabsf(apq) < 1e-20f) continue;
      const float tau = (Am[q][q] - Am[p][p]) / (2.0f * apq);
      const float tt = (tau >= 0.0f ? 1.0f : -1.0f) / (fabsf(tau) + sqrtf(1.0f + tau * tau));
      const float cc = 1.0f / sqrtf(1.0f + tt * tt);
      const float ss = tt * cc;
      for (int k = 0; k < 3; ++k) { const float kp = Am[k][p], kq = Am[k][q];
        Am[k][p] = cc * kp - ss * kq; Am[k][q] = ss * kp + cc * kq; }
      for (int k = 0; k < 3; ++k) { const float pk = Am[p][k], qk = Am[q][k];
        Am[p][k] = cc * pk - ss * qk; Am[q][k] = ss * pk + cc * qk; }
      for (int k = 0; k < 3; ++k) { const float kp = V[k][p], kq = V[k][q];
        V[k][p] = cc * kp - ss * kq; V[k][q] = ss * kp + cc * kq; }
    }
  }
  float lam[3] = {Am[0][0], Am[1][1], Am[2][2]};
  for (int i = 0; i < 2; ++i)
    for (int j = i + 1; j < 3; ++j)
      if (lam[j] > lam[i]) {
        const float tl = lam[i]; lam[i] = lam[j]; lam[j] = tl;
        for (int k = 0; k < 3; ++k) { const float tv = V[k][i]; V[k][i] = V[k][j]; V[k][j] = tv; }
      }

  // U columns = normalize(H * v_i); repair degenerate smallest column.
  float U[3][3];
  for (int i = 0; i < 3; ++i) {
    const float u0 = H[0][0]*V[0][i] + H[0][1]*V[1][i] + H[0][2]*V[2][i];
    const float u1 = H[1][0]*V[0][i] + H[1][1]*V[1][i] + H[1][2]*V[2][i];
    const float u2 = H[2][0]*V[0][i] + H[2][1]*V[1][i] + H[2][2]*V[2][i];
    const float nn = sqrtf(u0*u0 + u1*u1 + u2*u2);
    if (nn > 1e-12f) {
      const float inn = 1.0f / nn;
      U[0][i] = u0 * inn; U[1][i] = u1 * inn; U[2][i] = u2 * inn;
    } else if (i == 2) {
      const float x0 = U[1][0]*U[2][1] - U[2][0]*U[1][1];
      const float x1 = U[2][0]*U[0][1] - U[0][0]*U[2][1];
      const float x2 = U[0][0]*U[1][1] - U[1][0]*U[0][1];
      const float xn = sqrtf(x0*x0 + x1*x1 + x2*x2);
      const float ixn = (xn > 1e-12f) ? 1.0f / xn : 0.0f;
      U[0][2] = x0 * ixn; U[1][2] = x1 * ixn; U[2][2] = x2 * ixn;
    } else {
      U[0][i] = (i == 0) ? 1.0f : 0.0f;
      U[1][i] = (i == 1) ? 1.0f : 0.0f;
      U[2][i] = 0.0f;
    }
  }

  const float detU = U[0][0]*(U[1][1]*U[2][2]-U[1][2]*U[2][1])
                   - U[0][1]*(U[1][0]*U[2][2]-U[1][2]*U[2][0])
                   + U[0][2]*(U[1][0]*U[2][1]-U[1][1]*U[2][0]);
  const float detV = V[0][0]*(V[1][1]*V[2][2]-V[1][2]*V[2][1])
                   - V[0][1]*(V[1][0]*V[2][2]-V[1][2]*V[2][0])
                   + V[0][2]*(V[1][0]*V[2][1]-V[1][1]*V[2][0]);
  const float dd = (detU * detV < 0.0f) ? -1.0f : 1.0f;

  float R[3][3];
  for (int i = 0; i < 3; ++i)
    for (int j = 0; j < 3; ++j)
      R[i][j] = V[i][0]*U[j][0] + V[i][1]*U[j][1] + dd * V[i][2]*U[j][2];

  float t0 = c2[0] - (R[0][0]*c1[0] + R[0][1]*c1[1] + R[0][2]*c1[2]);
  float t1 = c2[1] - (R[1][0]*c1[0] + R[1][1]*c1[1] + R[1][2]*c1[2]);
  float t2 = c2[2] - (R[2][0]*c1[0] + R[2][1]*c1[1] + R[2][2]*c1[2]);
  const float tn = sqrtf(t0*t0 + t1*t1 + t2*t2);
  const float itn = 1.0f / fmaxf(tn, 1e-12f);
  t0 *= itn; t1 *= itn; t2 *= itn;

  if (Aa[16] < 4.5f) {  // count < 5 fallback
    for (int i = 0; i < 3; ++i)
      for (int j = 0; j < 3; ++j)
        R[i][j] = (i == j) ? 1.0f : 0.0f;
    t0 = 0.0f; t1 = 0.0f; t2 = 1.0f;
  }

  for (int i = 0; i < 3; ++i)
    for (int j = 0; j < 3; ++j)
      out[b * 9 + i * 3 + j] = R[i][j];
  out[NB * 9 + b * 3 + 0] = t0;
  out[NB * 9 + b * 3 + 1] = t1;
  out[NB * 9 + b * 3 + 2] = t2;
}

// ---------------------------------------------------------------------------
extern "C" void kernel_launch(void* const* d_in, const int* in_sizes, int n_in,
                              void* d_out, int out_size, void* d_ws, size_t ws_size,
                              hipStream_t stream) {
  const float* pos1   = (const float*)d_in[0];
  const float* pos2   = (const float*)d_in[1];
  const float* scores = (const float*)d_in[2];
  // d_in[3] = K is unused by the reference computation
  const float* w1 = (const float*)d_in[4];
  const float* b1 = (const float*)d_in[5];
  const float* w2 = (const float*)d_in[6];
  const float* b2 = (const float*)d_in[7];

  float* ws    = (float*)d_ws;
  float* max_s = ws + WS_MAXS;  // NB floats
  float* acc   = ws + WS_ACC;   // NB * ACC_STRIDE floats

  k_prepare<<<1, 64, 0, stream>>>(w1, b1, w2, b2, ws);
  k_max_init<<<NB, 256, 0, stream>>>(scores, max_s, acc);
  k_weights_cov<<<NB * 32 * 8, 32, 0, stream>>>(pos1, pos2, scores,
                                                max_s, ws, acc);
  k_finalize<<<1, 32, 0, stream>>>(acc, (float*)d_out);
}